// NystromformerAttention_46566035423449
// MI455X (gfx1250) — compile-verified
//
#include <hip/hip_runtime.h>
#include <hip/hip_bf16.h>
#include <math.h>

// ---- problem constants ----
#define B_    2
#define S_    4096
#define HID_  768
#define H_    12
#define D_    64
#define M_    64
#define SEG_  64
#define CK_   33
#define BH_   (B_ * H_)
#define LS_   68          // padded LDS row stride for Newton-Schulz matrices

typedef __attribute__((ext_vector_type(2))) float v2f;
typedef __attribute__((ext_vector_type(8))) float v8f;

// ---------------- fp32 WMMA 16x16x4 wave-level primitives (wave32) ----------
// A 16x4: lanes 0-15 -> M=0..15 with K={0,1}; lanes 16-31 -> same M, K={2,3}
// B 4x16: lanes 0-15 -> N=0..15 with K={0,1}; lanes 16-31 -> same N, K={2,3}
// C/D 16x16 f32: VGPR i: M=i (lanes 0-15), M=i+8 (lanes 16-31); N = lane&15

__device__ __forceinline__ v8f wmma4(v2f a, v2f b, v8f c) {
  return __builtin_amdgcn_wmma_f32_16x16x4_f32(false, a, false, b, (short)0, c,
                                               false, false);
}

__device__ __forceinline__ v2f load_a(const float* A, int lda) {
  int l = threadIdx.x & 31;
  int m = l & 15, k0 = (l >> 4) << 1;
  v2f a; a.x = A[m * lda + k0]; a.y = A[m * lda + k0 + 1];
  return a;
}
// logical B[k][n] = Bp[n*ldb + k]   (N-major storage, e.g. weight @ W^T)
__device__ __forceinline__ v2f load_bT(const float* Bp, int ldb) {
  int l = threadIdx.x & 31;
  int n = l & 15, k0 = (l >> 4) << 1;
  v2f b; b.x = Bp[n * ldb + k0]; b.y = Bp[n * ldb + k0 + 1];
  return b;
}
// logical B[k][n] = Bp[k*ldb + n]   (K-major storage)
__device__ __forceinline__ v2f load_bR(const float* Bp, int ldb) {
  int l = threadIdx.x & 31;
  int n = l & 15, k0 = (l >> 4) << 1;
  v2f b; b.x = Bp[k0 * ldb + n]; b.y = Bp[(k0 + 1) * ldb + n];
  return b;
}

// ---------------- 1) QKV projection: [8192,768] @ [768,2304]^T + bias -------
__global__ void qkv_gemm_k(const float* __restrict__ x, const float* __restrict__ w,
                           const float* __restrict__ bias,
                           float* __restrict__ Q, float* __restrict__ K,
                           float* __restrict__ V) {
  const int Nt = (3 * HID_) / 16;                 // 144
  int wave = threadIdx.x >> 5;                    // 8 waves / block
  int tile = blockIdx.x * 8 + wave;               // 512*144 tiles total
  int mt = tile / Nt, nt = tile - mt * Nt;
  const float* Ap = x + (size_t)mt * 16 * HID_;
  const float* Bp = w + (size_t)nt * 16 * HID_;   // w row-major [2304][768]
  v8f acc = {};
  for (int k = 0; k < HID_; k += 4) {
    v2f a = load_a(Ap + k, HID_);
    v2f b = load_bT(Bp + k, HID_);
    acc = wmma4(a, b, acc);
  }
  int l = threadIdx.x & 31;
  int n = l & 15, mh = (l >> 4) << 3;
  int c = nt * 16 + n;
  int which = c / HID_;
  int hid = c - which * HID_;
  int h = hid >> 6, d = hid & 63;
  float bv = bias[c];
  float* Out = (which == 0) ? Q : ((which == 1) ? K : V);
#pragma unroll
  for (int i = 0; i < 8; ++i) {
    int r = mt * 16 + mh + i;                     // r = b*S + s
    int bb = r >> 12, s = r & (S_ - 1);
    Out[(((size_t)bb * H_ + h) * S_ + s) * D_ + d] = acc[i] + bv;
  }
}

// ---------------- 2) landmark means over contiguous segments of 64 ----------
__global__ void landmark_k(const float* __restrict__ Q, const float* __restrict__ K,
                           float* __restrict__ Ql, float* __restrict__ Kl) {
  int idx = blockIdx.x;                           // bh*M + m
  int d = threadIdx.x;                            // 0..63
  int m = idx & (M_ - 1);
  int bh = idx >> 6;
  const float* qp = Q + ((size_t)bh * S_ + m * SEG_) * D_ + d;
  const float* kp = K + ((size_t)bh * S_ + m * SEG_) * D_ + d;
  float sq = 0.f, sk = 0.f;
  for (int i = 0; i < SEG_; ++i) { sq += qp[(size_t)i * D_]; sk += kp[(size_t)i * D_]; }
  Ql[(size_t)idx * D_ + d] = sq * (1.0f / SEG_);
  Kl[(size_t)idx * D_ + d] = sk * (1.0f / SEG_);
}

// ------- 3) fused A @ Kl^T * scale -> row softmax(64).  kernel_1 / kernel_2 -
// block = 128 threads = 4 waves (one per 16-wide N tile), 16 rows per block.
__global__ void attn_lm_softmax_k(const float* __restrict__ A,
                                  const float* __restrict__ Kl,
                                  float* __restrict__ out, int rows) {
  int bh = blockIdx.y;
  int rt = blockIdx.x;
  int wave = threadIdx.x >> 5;
  const float* Ap = A + ((size_t)bh * rows + rt * 16) * D_;
  const float* Bp = Kl + (size_t)bh * M_ * D_ + wave * 16 * D_;
  v8f acc = {};
  for (int k = 0; k < D_; k += 4) {
    v2f a = load_a(Ap + k, D_);
    v2f b = load_bT(Bp + k, D_);
    acc = wmma4(a, b, acc);
  }
  __shared__ float sc[16][M_ + 1];
  int l = threadIdx.x & 31;
  int n = (wave << 4) + (l & 15);
  int mh = (l >> 4) << 3;
  const float scl = 0.125f;                       // 1/sqrt(64)
#pragma unroll
  for (int i = 0; i < 8; ++i) sc[mh + i][n] = acc[i] * scl;
  __syncthreads();
  int r = threadIdx.x >> 3;                       // 16 rows, 8 threads per row
  int t8 = threadIdx.x & 7;
  float mx = -3.0e38f;
  for (int j = t8; j < M_; j += 8) mx = fmaxf(mx, sc[r][j]);
#pragma unroll
  for (int o = 4; o > 0; o >>= 1) mx = fmaxf(mx, __shfl_xor(mx, o, 8));
  float sum = 0.f;
  for (int j = t8; j < M_; j += 8) { float e = __expf(sc[r][j] - mx); sc[r][j] = e; sum += e; }
#pragma unroll
  for (int o = 4; o > 0; o >>= 1) sum += __shfl_xor(sum, o, 8);
  float inv = 1.0f / sum;
  float* op = out + ((size_t)bh * rows + rt * 16 + r) * M_;
  for (int j = t8; j < M_; j += 8) op[j] = sc[r][j] * inv;
}

// ---------------- 4) Newton-Schulz scale: global max col-sum / row-sum ------
__global__ void init_red_k(unsigned* red) { if (threadIdx.x < 2) red[threadIdx.x] = 0u; }

__global__ void k2_reduce_k(const float* __restrict__ K2, unsigned* __restrict__ red) {
  int bh = blockIdx.x, t = threadIdx.x;
  const float* p = K2 + (size_t)bh * M_ * M_;
  float cs = 0.f, rs = 0.f;
  for (int i = 0; i < M_; ++i) { cs += fabsf(p[i * M_ + t]); rs += fabsf(p[t * M_ + i]); }
  __shared__ float scs[M_], srs[M_];
  scs[t] = cs; srs[t] = rs;
  __syncthreads();
  if (t == 0) {
    float m1 = 0.f, m2 = 0.f;
    for (int i = 0; i < M_; ++i) { m1 = fmaxf(m1, scs[i]); m2 = fmaxf(m2, srs[i]); }
    atomicMax(&red[0], __float_as_uint(m1));      // positive floats: uint order ok
    atomicMax(&red[1], __float_as_uint(m2));
  }
}

// ---------------- 5) Newton-Schulz pseudo-inverse, all in LDS ---------------
// C = cs * ( A @ (dg*I + bs*B) ), 16 waves each own one 16x16 tile of 64x64.
__device__ __forceinline__ void lds_mm(const float* A, const float* Bm, float* C,
                                       float dg, float bs, float cs) {
  int wave = threadIdx.x >> 5;
  int mt = wave >> 2, nt = wave & 3;
  int l = threadIdx.x & 31;
  int n = (nt << 4) + (l & 15);
  const float* Ap = A + mt * 16 * LS_;
  v8f acc = {};
  for (int k = 0; k < M_; k += 4) {
    v2f a = load_a(Ap + k, LS_);
    int k0 = k + ((l >> 4) << 1);
    v2f b;
    b.x = dg * ((k0     == n) ? 1.0f : 0.0f) + bs * Bm[k0 * LS_ + n];
    b.y = dg * ((k0 + 1 == n) ? 1.0f : 0.0f) + bs * Bm[(k0 + 1) * LS_ + n];
    acc = wmma4(a, b, acc);
  }
  int mh = (mt << 4) + ((l >> 4) << 3);
#pragma unroll
  for (int i = 0; i < 8; ++i) C[(mh + i) * LS_ + n] = cs * acc[i];
}

__global__ void newton_inv_k(const float* __restrict__ K2,
                             const unsigned* __restrict__ red,
                             float* __restrict__ invOut) {
  extern __shared__ float lds[];
  float* Km = lds;
  float* Vm = Km + M_ * LS_;
  float* T1 = Vm + M_ * LS_;
  float* T2 = T1 + M_ * LS_;
  float* T3 = T2 + M_ * LS_;
  int bh = blockIdx.x;
  const float* p = K2 + (size_t)bh * M_ * M_;
  float scale = 1.0f / (__uint_as_float(red[0]) * __uint_as_float(red[1]));
  for (int i = threadIdx.x; i < M_ * M_; i += blockDim.x) {
    int r = i >> 6, c = i & 63;
    float v = p[i];
    Km[r * LS_ + c] = v;
    Vm[c * LS_ + r] = v * scale;                  // V0 = scale * K^T
  }
  __syncthreads();
  for (int it = 0; it < 6; ++it) {
    lds_mm(Km, Vm, T1, 0.f, 1.f, 1.f);   __syncthreads();  // T1 = K@V
    lds_mm(T1, T1, T2, 7.f, -1.f, 1.f);  __syncthreads();  // T2 = KV@(7I-KV)
    lds_mm(T1, T2, T3, 15.f, -1.f, 1.f); __syncthreads();  // T3 = KV@(15I-T2)
    lds_mm(Vm, T3, T1, 13.f, -1.f, 0.25f); __syncthreads(); // T1 = .25*V@(13I-T3)
    for (int i = threadIdx.x; i < M_ * M_; i += blockDim.x) {
      int r = i >> 6, c = i & 63;
      Vm[r * LS_ + c] = T1[r * LS_ + c];
    }
    __syncthreads();
  }
  for (int i = threadIdx.x; i < M_ * M_; i += blockDim.x) {
    int r = i >> 6, c = i & 63;
    invOut[(size_t)bh * M_ * M_ + i] = Vm[r * LS_ + c];
  }
}

// ---------------- 6) kernel_3 scores: Ql @ K^T * scale ----------------------
__global__ void k3_gemm_k(const float* __restrict__ Ql, const float* __restrict__ K,
                          float* __restrict__ S3) {
  int bh = blockIdx.y;
  int wave = threadIdx.x >> 5;
  int tile = blockIdx.x * 16 + wave;              // 4 mt x 256 nt = 1024 tiles
  int mt = tile >> 8, nt = tile & 255;
  const float* Ap = Ql + (size_t)bh * M_ * D_ + mt * 16 * D_;
  const float* Bp = K + ((size_t)bh * S_ + nt * 16) * D_;
  v8f acc = {};
  for (int k = 0; k < D_; k += 4) {
    v2f a = load_a(Ap + k, D_);
    v2f b = load_bT(Bp + k, D_);
    acc = wmma4(a, b, acc);
  }
  int l = threadIdx.x & 31;
  int n = nt * 16 + (l & 15);
  int mh = mt * 16 + ((l >> 4) << 3);
#pragma unroll
  for (int i = 0; i < 8; ++i)
    S3[((size_t)bh * M_ + mh + i) * S_ + n] = acc[i] * 0.125f;
}

// ---------------- 7) in-place softmax over rows of length n (4096) ----------
__global__ void softmax_rows_k(float* __restrict__ P, int n) {
  size_t row = blockIdx.x;
  float* p = P + row * (size_t)n;
  int t = threadIdx.x;
  __shared__ float s[256];
  float mx = -3.0e38f;
  for (int j = t; j < n; j += 256) mx = fmaxf(mx, p[j]);
  s[t] = mx; __syncthreads();
  for (int o = 128; o > 0; o >>= 1) { if (t < o) s[t] = fmaxf(s[t], s[t + o]); __syncthreads(); }
  mx = s[0]; __syncthreads();
  float sum = 0.f;
  for (int j = t; j < n; j += 256) { float e = __expf(p[j] - mx); p[j] = e; sum += e; }
  s[t] = sum; __syncthreads();
  for (int o = 128; o > 0; o >>= 1) { if (t < o) s[t] += s[t + o]; __syncthreads(); }
  float inv = 1.0f / s[0];
  for (int j = t; j < n; j += 256) p[j] *= inv;
}

// ---------------- 8) W2 = kernel_3 @ V  ([64,4096]@[4096,64] per head) ------
__global__ void k3v_gemm_k(const float* __restrict__ K3, const float* __restrict__ V,
                           float* __restrict__ W2) {
  int bh = blockIdx.x;
  int wave = threadIdx.x >> 5;
  int mt = wave >> 2, nt = wave & 3;
  const float* Ap = K3 + (size_t)bh * M_ * S_ + (size_t)mt * 16 * S_;
  const float* Bp = V + (size_t)bh * S_ * D_ + nt * 16;
  v8f acc = {};
  for (int k = 0; k < S_; k += 4) {
    v2f a = load_a(Ap + k, S_);
    v2f b = load_bR(Bp + (size_t)k * D_, D_);
    acc = wmma4(a, b, acc);
  }
  int l = threadIdx.x & 31;
  int n = nt * 16 + (l & 15);
  int mh = mt * 16 + ((l >> 4) << 3);
#pragma unroll
  for (int i = 0; i < 8; ++i) W2[((size_t)bh * M_ + mh + i) * D_ + n] = acc[i];
}

// ---------------- 9) W1 = inv @ W2 ------------------------------------------
__global__ void w1_gemm_k(const float* __restrict__ INVm, const float* __restrict__ W2,
                          float* __restrict__ W1) {
  int bh = blockIdx.x;
  int wave = threadIdx.x >> 5;
  int mt = wave >> 2, nt = wave & 3;
  const float* Ap = INVm + (size_t)bh * M_ * M_ + mt * 16 * M_;
  const float* Bp = W2 + (size_t)bh * M_ * D_ + nt * 16;
  v8f acc = {};
  for (int k = 0; k < M_; k += 4) {
    v2f a = load_a(Ap + k, M_);
    v2f b = load_bR(Bp + k * D_, D_);
    acc = wmma4(a, b, acc);
  }
  int l = threadIdx.x & 31;
  int n = nt * 16 + (l & 15);
  int mh = mt * 16 + ((l >> 4) << 3);
#pragma unroll
  for (int i = 0; i < 8; ++i) W1[((size_t)bh * M_ + mh + i) * D_ + n] = acc[i];
}

// ---------------- 10) X = kernel_1 @ W1 -------------------------------------
__global__ void x_gemm_k(const float* __restrict__ K1, const float* __restrict__ W1,
                         float* __restrict__ X) {
  int bh = blockIdx.y;
  int rt = blockIdx.x;
  int wave = threadIdx.x >> 5;                    // nt 0..3
  const float* Ap = K1 + ((size_t)bh * S_ + rt * 16) * M_;
  const float* Bp = W1 + (size_t)bh * M_ * D_ + wave * 16;
  v8f acc = {};
  for (int k = 0; k < M_; k += 4) {
    v2f a = load_a(Ap + k, M_);
    v2f b = load_bR(Bp + k * D_, D_);
    acc = wmma4(a, b, acc);
  }
  int l = threadIdx.x & 31;
  int n = wave * 16 + (l & 15);
  int mh = (l >> 4) << 3;
#pragma unroll
  for (int i = 0; i < 8; ++i)
    X[((size_t)bh * S_ + rt * 16 + mh + i) * D_ + n] = acc[i];
}

// ---------------- 11) depthwise conv over S, added into X -------------------
__global__ void conv_add_k(float* __restrict__ X, const float* __restrict__ V,
                           const float* __restrict__ cw) {
  size_t i = (size_t)blockIdx.x * 256 + threadIdx.x;   // index into [bh][s][d]
  int d = (int)(i & 63);
  size_t rest = i >> 6;
  int s = (int)(rest & (S_ - 1));
  int bh = (int)(rest >> 12);
  int h = bh % H_;
  const float* vp = V + (size_t)bh * S_ * D_ + d;
  float acc = 0.f;
#pragma unroll
  for (int t = 0; t < CK_; ++t) {
    int ss = s + t - (CK_ / 2);
    if (ss >= 0 && ss < S_) acc += vp[(size_t)ss * D_] * cw[h * CK_ + t];
  }
  X[i] += acc;
}

// ---------------- 12) output projection: gather heads + @ out_w^T + bias ----
__global__ void out_gemm_k(const float* __restrict__ X, const float* __restrict__ w,
                           const float* __restrict__ bias, float* __restrict__ out) {
  const int Nt = HID_ / 16;                       // 48
  int wave = threadIdx.x >> 5;
  int tile = blockIdx.x * 8 + wave;               // 512*48 tiles
  int mt = tile / Nt, nt = tile - mt * Nt;
  int l = threadIdx.x & 31;
  int m = l & 15;
  int r = mt * 16 + m;                            // r = b*S + s
  int bb = r >> 12, s = r & (S_ - 1);
  const float* Bp = w + (size_t)nt * 16 * HID_;
  v8f acc = {};
  for (int k = 0; k < HID_; k += 4) {
    int k0 = k + ((l >> 4) << 1);
    int h = k0 >> 6, d = k0 & 63;                 // k0 even -> pair stays in head
    const float* ap = X + (((size_t)bb * H_ + h) * S_ + s) * D_ + d;
    v2f a; a.x = ap[0]; a.y = ap[1];
    v2f b = load_bT(Bp + k, HID_);
    acc = wmma4(a, b, acc);
  }
  int n = l & 15, mh = (l >> 4) << 3;
  int c = nt * 16 + n;
  float bv = bias[c];
#pragma unroll
  for (int i = 0; i < 8; ++i) {
    int rr = mt * 16 + mh + i;
    out[(size_t)rr * HID_ + c] = acc[i] + bv;
  }
}

// ---------------- launch ----------------------------------------------------
extern "C" void kernel_launch(void* const* d_in, const int* in_sizes, int n_in,
                              void* d_out, int out_size, void* d_ws, size_t ws_size,
                              hipStream_t stream) {
  const float* x      = (const float*)d_in[0];
  const float* qkv_w  = (const float*)d_in[1];
  const float* qkv_b  = (const float*)d_in[2];
  const float* conv_w = (const float*)d_in[3];
  const float* out_w  = (const float*)d_in[4];
  const float* out_b  = (const float*)d_in[5];
  float* out = (float*)d_out;

  float* ws = (float*)d_ws;
  const size_t NBH     = (size_t)BH_;             // 24
  const size_t SZ_BHSD = NBH * S_ * D_;           // 6.29M floats
  const size_t SZ_BHMD = NBH * M_ * D_;           // 98304 floats
  size_t o = 0;
  float* Q    = ws + o; o += SZ_BHSD;
  float* K    = ws + o; o += SZ_BHSD;
  float* V    = ws + o; o += SZ_BHSD;
  float* K1   = ws + o; o += SZ_BHSD;             // [bh,S,M]
  float* K3   = ws + o; o += SZ_BHSD;             // [bh,M,S]
  float* X    = ws + o; o += SZ_BHSD;
  float* Ql   = ws + o; o += SZ_BHMD;
  float* Kl   = ws + o; o += SZ_BHMD;
  float* K2   = ws + o; o += SZ_BHMD;
  float* INVm = ws + o; o += SZ_BHMD;
  float* W2   = ws + o; o += SZ_BHMD;
  float* W1   = ws + o; o += SZ_BHMD;
  unsigned* red = (unsigned*)(ws + o); o += 64;   // total ~153 MB of d_ws

  init_red_k<<<dim3(1), dim3(32), 0, stream>>>(red);
  qkv_gemm_k<<<dim3((512 * 144) / 8), dim3(256), 0, stream>>>(x, qkv_w, qkv_b, Q, K, V);
  landmark_k<<<dim3((unsigned)(NBH * M_)), dim3(64), 0, stream>>>(Q, K, Ql, Kl);
  attn_lm_softmax_k<<<dim3(S_ / 16, (unsigned)NBH), dim3(128), 0, stream>>>(Q, Kl, K1, S_);
  attn_lm_softmax_k<<<dim3(M_ / 16, (unsigned)NBH), dim3(128), 0, stream>>>(Ql, Kl, K2, M_);
  k2_reduce_k<<<dim3((unsigned)NBH), dim3(64), 0, stream>>>(K2, red);
  newton_inv_k<<<dim3((unsigned)NBH), dim3(512), 5 * M_ * LS_ * sizeof(float), stream>>>(K2, red, INVm);
  k3_gemm_k<<<dim3(64, (unsigned)NBH), dim3(512), 0, stream>>>(Ql, K, K3);
  softmax_rows_k<<<dim3((unsigned)(NBH * M_)), dim3(256), 0, stream>>>(K3, S_);
  k3v_gemm_k<<<dim3((unsigned)NBH), dim3(512), 0, stream>>>(K3, V, W2);
  w1_gemm_k<<<dim3((unsigned)NBH), dim3(512), 0, stream>>>(INVm, W2, W1);
  x_gemm_k<<<dim3(S_ / 16, (unsigned)NBH), dim3(128), 0, stream>>>(K1, W1, X);
  conv_add_k<<<dim3((unsigned)(SZ_BHSD / 256)), dim3(256), 0, stream>>>(X, V, conv_w);
  out_gemm_k<<<dim3((512 * 48) / 8), dim3(256), 0, stream>>>(X, out_w, out_b, out);
}